// IGCNet_23536420782218
// MI455X (gfx1250) — compile-verified
//
#include <hip/hip_runtime.h>

#define N_NODES_C 100000
#define N_EDGES_C 3200000

typedef __attribute__((ext_vector_type(2))) float v2f;
typedef __attribute__((ext_vector_type(8))) float v8f;

// D = A(16x4 f32) * B(4x16 f32) + C(16x16 f32), fp32 WMMA, K=4 per step.
__device__ __forceinline__ v8f wmma4(v2f a, v2f b, v8f c) {
    return __builtin_amdgcn_wmma_f32_16x16x4_f32(
        /*neg_a=*/false, a, /*neg_b=*/false, b,
        /*c_mod=*/(short)0, c, /*reuse_a=*/false, /*reuse_b=*/false);
}

// ---------------- agg init: agg[N,4] = 0 ----------------
__global__ void __launch_bounds__(256) igc_init_agg(unsigned int* agg, int n) {
    int i = blockIdx.x * blockDim.x + threadIdx.x;
    if (i < n) agg[i] = 0u;  // 0x0 == +0.0f
}

// ---------------- edge phase: msg = relu([x[src], ea] @ W1 + b1); agg = max ----------------
// One wave = 16 edges. A: 16x24 fp32 (6 K=4 steps). B: W1 columns 0..3, rest 0.
// Scatter: msg -> LDS [n][m] transpose -> 64 atomic ops issued as 2 full-EXEC atomics.
// edge_index is int64 but ids < 2^31: read low dwords only (little-endian) -> 32-bit paths.
__global__ void __launch_bounds__(256) igc_edge(const float* __restrict__ x,
                                                const float* __restrict__ ea,
                                                const int* __restrict__ ei32,  // int64 viewed as int pairs
                                                const float* __restrict__ W1,
                                                const float* __restrict__ b1,
                                                unsigned int* __restrict__ agg,
                                                int E) {
    __shared__ float mbuf[8][64];        // per-wave msg staging, [n][m] layout
    const int lane = threadIdx.x & 31;
    const int half = lane >> 4;          // 0: K lanes {0,1}, 1: K lanes {2,3}
    const int n    = lane & 15;          // A row (edge-in-tile); B/C/D column (valid n<4)
    const int w    = threadIdx.x >> 5;
    const int wavesPerBlock = blockDim.x >> 5;
    const int waveId = blockIdx.x * wavesPerBlock + w;
    const int nWaves = gridDim.x * wavesPerBlock;

    // Scatter-phase mapping: op o = (m, nn); lane handles o=lane (m1) and o=lane+32 (m1+8).
    const int nn = lane & 3;
    const int m1 = lane >> 2;            // 0..7

    // Hoist B (W1 in [24,4] row-major) and bias into registers.
    v2f B[6];
#pragma unroll
    for (int s = 0; s < 6; ++s) {
        int k0 = 4 * s + 2 * half;
        v2f b; b.x = 0.f; b.y = 0.f;
        if (n < 4) { b.x = W1[k0 * 4 + n]; b.y = W1[(k0 + 1) * 4 + n]; }
        B[s] = b;
    }
    const float cinit = (n < 4) ? b1[n] : 0.f;

    const int nTiles = (E + 15) >> 4;
    for (int t = waveId; t < nTiles; t += nWaves) {
        const int  ebase    = __builtin_amdgcn_readfirstlane(t) << 4;  // wave-uniform (scalar)
        const bool fullTile = (ebase + 16) <= E;                        // scalar branch
        const int  erow     = ebase + n;             // this lane's A row (edge)
        const int  erc      = erow < E ? erow : 0;
        const int  s0  = ei32[2 * erc];              // src node (low dword of int64)
        const int  dme = ei32[2 * (E + erc)];        // dst node (low dword of int64)
        const float* xr = x  + s0 * 16;
        const float* er = ea + erc * 8;

        v2f A[6];
#pragma unroll
        for (int s = 0; s < 4; ++s) {                // K 0..15 from x[src]
            int k0 = 4 * s + 2 * half;               // even -> 8B aligned
            A[s] = *(const v2f*)(xr + k0);
        }
#pragma unroll
        for (int s = 4; s < 6; ++s) {                // K 16..23 from edge_attr
            int k0 = 4 * (s - 4) + 2 * half;
            A[s] = *(const v2f*)(er + k0);
        }

        v8f C;
#pragma unroll
        for (int r = 0; r < 8; ++r) C[r] = cinit;
#pragma unroll
        for (int s = 0; s < 6; ++s) C = wmma4(A[s], B[s], C);

        // relu(msg) -> LDS transposed [n][m]; C[r] @ lane = msg[m = r + 8*half][n]
        if (n < 4) {
            float* p = &mbuf[w][n * 16 + 8 * half];  // consecutive in r -> b128 stores
#pragma unroll
            for (int r = 0; r < 8; r += 2) {
                v2f v;
                v.x = fmaxf(C[r], 0.f);
                v.y = fmaxf(C[r + 1], 0.f);
                *(v2f*)(p + r) = v;
            }
        }
        asm volatile("s_wait_dscnt 0x0" ::: "memory");   // same-wave LDS RAW fence

        // Each lane: 2 of the 64 scatter ops; 32-bit dst via single bpermute each.
        const float v1 = mbuf[w][nn * 16 + m1];
        const float v2 = mbuf[w][nn * 16 + m1 + 8];
        const int d1 = __shfl(dme, m1, 32);
        const int d2 = __shfl(dme, m1 + 8, 32);
        if (fullTile) {
            atomicMax(&agg[d1 * 4 + nn], __float_as_uint(v1));
            atomicMax(&agg[d2 * 4 + nn], __float_as_uint(v2));
        } else {
            if (ebase + m1 < E)     atomicMax(&agg[d1 * 4 + nn], __float_as_uint(v1));
            if (ebase + m1 + 8 < E) atomicMax(&agg[d2 * 4 + nn], __float_as_uint(v2));
        }
    }
}

// ---------------- node phase: h=relu([x,agg]@W2a+b2a); comb=h@W2b+b2b; normalize; concat ----------------
__global__ void __launch_bounds__(256) igc_node(const float* __restrict__ xin,
                                                const float* __restrict__ agg,
                                                const float* __restrict__ W2a,
                                                const float* __restrict__ b2a,
                                                const float* __restrict__ W2b,
                                                const float* __restrict__ b2b,
                                                float* __restrict__ xout,
                                                int N) {
    __shared__ float hbuf[8][64];                    // per-wave 16x4 h staging
    const int lane = threadIdx.x & 31;
    const int half = lane >> 4;
    const int n    = lane & 15;
    const int w    = threadIdx.x >> 5;
    const int wavesPerBlock = blockDim.x >> 5;
    const int waveId = blockIdx.x * wavesPerBlock + w;
    const int nWaves = gridDim.x * wavesPerBlock;

    // Stage a weights: W2a [20,4]
    v2f Ba[5];
#pragma unroll
    for (int s = 0; s < 5; ++s) {
        int k0 = 4 * s + 2 * half;
        v2f b; b.x = 0.f; b.y = 0.f;
        if (n < 4) { b.x = W2a[k0 * 4 + n]; b.y = W2a[(k0 + 1) * 4 + n]; }
        Ba[s] = b;
    }
    const float ca = (n < 4) ? b2a[n] : 0.f;
    // Stage b weights: W2b [4,8]
    v2f Bb; Bb.x = 0.f; Bb.y = 0.f;
    {
        int k0 = 2 * half;
        if (n < 8) { Bb.x = W2b[k0 * 8 + n]; Bb.y = W2b[(k0 + 1) * 8 + n]; }
    }
    const float cb = (n < 8) ? b2b[n] : 0.f;

    const int nTiles = (N + 15) >> 4;
    for (int t = waveId; t < nTiles; t += nWaves) {
        const int  base  = __builtin_amdgcn_readfirstlane(t) << 4;  // wave-uniform (scalar)
        const bool fullT = (base + 16) <= N;
        const int  row   = base + n;
        const int  rc    = row < N ? row : 0;
        const float* xr  = xin + rc * 16;

        // Stage a: A = [x (K=0..15) | agg (K=16..19)]
        v2f A[5];
#pragma unroll
        for (int s = 0; s < 4; ++s) A[s] = *(const v2f*)(xr + 4 * s + 2 * half);
        A[4] = *(const v2f*)(agg + rc * 4 + 2 * half);

        v8f C;
#pragma unroll
        for (int r = 0; r < 8; ++r) C[r] = ca;
#pragma unroll
        for (int s = 0; s < 5; ++s) C = wmma4(A[s], Ba[s], C);

        // relu(h) -> LDS in row-major [m][k] for stage-b A layout
        if (n < 4) {
#pragma unroll
            for (int r = 0; r < 8; ++r) {
                const int m = r + 8 * half;
                hbuf[w][m * 4 + n] = fmaxf(C[r], 0.f);
            }
        }
        asm volatile("s_wait_dscnt 0x0" ::: "memory");   // same-wave LDS RAW fence

        v2f Ab; Ab.x = hbuf[w][n * 4 + 2 * half];
                Ab.y = hbuf[w][n * 4 + 2 * half + 1];

        v8f D;
#pragma unroll
        for (int r = 0; r < 8; ++r) D[r] = cb;
        D = wmma4(Ab, Bb, D);

        // Unit-ball normalize per node: valid cols n=0..7 live in 8-lane groups.
#pragma unroll
        for (int r = 0; r < 8; ++r) {
            float v = D[r];
            float s = v * v;
            s += __shfl_xor(s, 1, 32);
            s += __shfl_xor(s, 2, 32);
            s += __shfl_xor(s, 4, 32);
            float nor = sqrtf(s);
            float scale = (nor > 1.f) ? (1.f / nor) : 1.f;
            D[r] = v * scale;
        }

        // out[node] = [comb(8), xin[node][0:8]]
        if (n < 8) {
            if (fullT) {
#pragma unroll
                for (int r = 0; r < 8; ++r) {
                    const int node = base + r + 8 * half;
                    xout[node * 16 + n] = D[r];
                    xout[node * 16 + 8 + n] = xin[node * 16 + n];
                }
            } else {
#pragma unroll
                for (int r = 0; r < 8; ++r) {
                    const int node = base + r + 8 * half;
                    if (node < N) {
                        xout[node * 16 + n] = D[r];
                        xout[node * 16 + 8 + n] = xin[node * 16 + n];
                    }
                }
            }
        }
    }
}

extern "C" void kernel_launch(void* const* d_in, const int* in_sizes, int n_in,
                              void* d_out, int out_size, void* d_ws, size_t ws_size,
                              hipStream_t stream) {
    const float* x   = (const float*)d_in[0];
    const float* ea  = (const float*)d_in[1];
    const int*   ei32 = (const int*)d_in[2];   // int64 edge_index viewed as int pairs (ids < 2^31)
    const float* W1  = (const float*)d_in[3];
    const float* b1  = (const float*)d_in[4];
    const float* W2a = (const float*)d_in[5];
    const float* b2a = (const float*)d_in[6];
    const float* W2b = (const float*)d_in[7];
    const float* b2b = (const float*)d_in[8];
    float* out = (float*)d_out;

    float* wsf = (float*)d_ws;
    float* buf1 = wsf;
    float* buf2 = wsf + (size_t)N_NODES_C * 16;
    unsigned int* agg = (unsigned int*)(wsf + 2 * (size_t)N_NODES_C * 16);

    const float* cur = x;
    float* nxt[3] = {buf1, buf2, out};
    for (int r = 0; r < 3; ++r) {
        igc_init_agg<<<(N_NODES_C * 4 + 255) / 256, 256, 0, stream>>>(agg, N_NODES_C * 4);
        igc_edge<<<1024, 256, 0, stream>>>(cur, ea, ei32, W1, b1, agg, N_EDGES_C);
        igc_node<<<256, 256, 0, stream>>>(cur, (const float*)agg, W2a, b2a, W2b, b2b, nxt[r], N_NODES_C);
        cur = nxt[r];
    }
}